// LabelSizeFilter_36876589203620
// MI455X (gfx1250) — compile-verified
//
#include <hip/hip_runtime.h>

#define NUM_LABELS   2048
#define MIN_SZ       10000u
#define MAX_SZ       20000u

typedef __attribute__((ext_vector_type(4))) int   v4i;
typedef __attribute__((ext_vector_type(4))) float v4f;

#define AS1 __attribute__((address_space(1)))
#define AS3 __attribute__((address_space(3)))

// ---------------------------------------------------------------------------
// Kernel 1: zero the global histogram (d_ws is poisoned, must re-zero per call)
// ---------------------------------------------------------------------------
__global__ void LSF_zero_hist(unsigned int* __restrict__ hist) {
    int i = blockIdx.x * blockDim.x + threadIdx.x;
    if (i < NUM_LABELS) hist[i] = 0u;
}

// ---------------------------------------------------------------------------
// Kernel 2: LDS-privatized histogram. 2048 bins * 4B = 8KB LDS per workgroup
// (WGP has 320KB, so occupancy is unconstrained). ds_add_u32 per pixel,
// one global_atomic_add per non-zero bin per block at the end.
// Image is read with regular-temporal b128 loads so it stays resident in the
// 192MB L2 for the remap pass.
// ---------------------------------------------------------------------------
__global__ void LSF_hist(const int* __restrict__ img,
                         unsigned int* __restrict__ hist, int n4) {
    __shared__ unsigned int sh[NUM_LABELS];
    for (int i = threadIdx.x; i < NUM_LABELS; i += blockDim.x) sh[i] = 0u;
    __syncthreads();

    const v4i* __restrict__ img4 = (const v4i*)img;
    int stride = gridDim.x * blockDim.x;
    for (int i = blockIdx.x * blockDim.x + threadIdx.x; i < n4; i += stride) {
        v4i v = img4[i];                       // global_load_b128
        atomicAdd(&sh[v.x & (NUM_LABELS - 1)], 1u);   // ds_add_u32
        atomicAdd(&sh[v.y & (NUM_LABELS - 1)], 1u);
        atomicAdd(&sh[v.z & (NUM_LABELS - 1)], 1u);
        atomicAdd(&sh[v.w & (NUM_LABELS - 1)], 1u);
    }
    __syncthreads();

    for (int i = threadIdx.x; i < NUM_LABELS; i += blockDim.x) {
        unsigned int c = sh[i];
        if (c) atomicAdd(&hist[i], c);         // global_atomic_add_u32
    }
}

// ---------------------------------------------------------------------------
// Kernel 3: build the remap table. Output dtype is float per harness contract.
// ---------------------------------------------------------------------------
__global__ void LSF_build_remap(const unsigned int* __restrict__ hist,
                                float* __restrict__ remapf) {
    int i = blockIdx.x * blockDim.x + threadIdx.x;
    if (i < NUM_LABELS) {
        unsigned int c = hist[i];
        bool bad = ((c < MIN_SZ) | (c > MAX_SZ)) && (i != 0);
        remapf[i] = bad ? 0.0f : (float)i;
    }
}

// ---------------------------------------------------------------------------
// Kernel 4: remap gather. Broadcast the 8KB remap table into LDS via the
// CDNA5 async global->LDS path (ASYNCcnt), then gather from LDS (ds_load)
// and write the output with non-temporal b128 stores (output is never
// re-read; NT keeps the image resident in L2 across graph replays).
// ---------------------------------------------------------------------------
__global__ void LSF_apply(const int* __restrict__ img,
                          const float* __restrict__ remapf,
                          float* __restrict__ out, int n4) {
    __shared__ float sh[NUM_LABELS];

#if defined(__AMDGCN__) && __has_builtin(__builtin_amdgcn_global_load_async_to_lds_b32)
    {
        // gfx1250 async path: GLOBAL_LOAD_ASYNC_TO_LDS_B32 (ASYNCcnt-tracked).
        // Builtin signature wants (AS1 int*, AS3 int*, imm offset, imm cpol).
        AS3 int* lp = (AS3 int*)sh;
        AS1 int* gp = (AS1 int*)remapf;
        for (int i = threadIdx.x; i < NUM_LABELS; i += blockDim.x) {
            __builtin_amdgcn_global_load_async_to_lds_b32(
                gp + i, lp + i, /*imm offset*/0, /*cpol*/0);
        }
#if __has_builtin(__builtin_amdgcn_s_wait_asynccnt)
        __builtin_amdgcn_s_wait_asynccnt(0);
#else
        asm volatile("s_wait_asynccnt 0" ::: "memory");
#endif
    }
#else
    for (int i = threadIdx.x; i < NUM_LABELS; i += blockDim.x) sh[i] = remapf[i];
#endif
    __syncthreads();

    const v4i* __restrict__ img4 = (const v4i*)img;
    v4f*       __restrict__ out4 = (v4f*)out;
    int stride = gridDim.x * blockDim.x;
    for (int i = blockIdx.x * blockDim.x + threadIdx.x; i < n4; i += stride) {
        v4i v = img4[i];                       // global_load_b128 (L2-resident)
        v4f r;
        r.x = sh[v.x & (NUM_LABELS - 1)];      // ds_load_b32 gathers
        r.y = sh[v.y & (NUM_LABELS - 1)];
        r.z = sh[v.z & (NUM_LABELS - 1)];
        r.w = sh[v.w & (NUM_LABELS - 1)];
        __builtin_nontemporal_store(r, &out4[i]);  // global_store_b128 nt
    }
}

// ---------------------------------------------------------------------------
extern "C" void kernel_launch(void* const* d_in, const int* in_sizes, int n_in,
                              void* d_out, int out_size, void* d_ws, size_t ws_size,
                              hipStream_t stream) {
    const int* img = (const int*)d_in[0];
    int n  = in_sizes[0];          // 32*1024*1024
    int n4 = n >> 2;               // image length is a multiple of 4

    unsigned int* hist   = (unsigned int*)d_ws;
    float*        remapf = (float*)((char*)d_ws + NUM_LABELS * sizeof(unsigned int));
    float*        out    = (float*)d_out;

    LSF_zero_hist<<<(NUM_LABELS + 255) / 256, 256, 0, stream>>>(hist);
    LSF_hist<<<2048, 256, 0, stream>>>(img, hist, n4);
    LSF_build_remap<<<(NUM_LABELS + 255) / 256, 256, 0, stream>>>(hist, remapf);
    LSF_apply<<<4096, 256, 0, stream>>>(img, remapf, out, n4);
}